// MAMGCN_submodule_54941221651372
// MI455X (gfx1250) — compile-verified
//
#include <hip/hip_runtime.h>
#include <hip/hip_bf16.h>

#define B_    16
#define N_    1024
#define FIN_  64
#define T_    12
#define K_    3
#define FOUT_ 64
#define FT_   (FIN_ * T_)   // 768

typedef float v2f __attribute__((ext_vector_type(2)));
typedef float v8f __attribute__((ext_vector_type(8)));

__device__ __forceinline__ v8f wmma_f32_16x16x4(v2f a, v2f b, v8f c) {
    // D = A(16x4 f32) x B(4x16 f32) + C(16x16 f32)
    return __builtin_amdgcn_wmma_f32_16x16x4_f32(
        /*neg_a=*/false, a, /*neg_b=*/false, b,
        /*c_mod=*/(short)0, c, /*reuse_a=*/false, /*reuse_b=*/false);
}

// ---------------------------------------------------------------------------
// Stage 1: lhs[b,n,t'] = sum_f (sum_t x[b,n,f,t]*W1[t]) * W2[f,t']
//          rhs[b,n,t]  = sum_f W3[f]*x[b,n,f,t]
// One wave per (b,n); lane handles f = {2*lane, 2*lane+1}; cross-lane reduce.
// ---------------------------------------------------------------------------
__global__ void k_attn_vecs(const float* __restrict__ x,
                            const float* __restrict__ W1,
                            const float* __restrict__ W2,
                            const float* __restrict__ W3,
                            float* __restrict__ lhs,
                            float* __restrict__ rhs)
{
    const int lane = threadIdx.x & 31;
    const int wid  = blockIdx.x * 8 + (threadIdx.x >> 5);   // b*N + n, < 16384

    const float4* xp4 = (const float4*)(x + (size_t)wid * FT_ + lane * 24);
    float xv[24];
#pragma unroll
    for (int i = 0; i < 6; ++i) {
        float4 v = xp4[i];
        xv[4*i+0] = v.x; xv[4*i+1] = v.y; xv[4*i+2] = v.z; xv[4*i+3] = v.w;
    }
    float l1a = 0.f, l1b = 0.f;
#pragma unroll
    for (int t = 0; t < T_; ++t) {
        const float w = W1[t];
        l1a += xv[t] * w;
        l1b += xv[T_ + t] * w;
    }
    const int f0 = 2 * lane, f1 = 2 * lane + 1;
    const float w3a = W3[f0], w3b = W3[f1];
    float aL[T_], aR[T_];
#pragma unroll
    for (int t = 0; t < T_; ++t) {
        aL[t] = l1a * W2[f0 * T_ + t] + l1b * W2[f1 * T_ + t];
        aR[t] = w3a * xv[t] + w3b * xv[T_ + t];
    }
#pragma unroll
    for (int t = 0; t < T_; ++t) {
        float v = aL[t], w = aR[t];
#pragma unroll
        for (int off = 16; off; off >>= 1) {
            v += __shfl_xor(v, off, 32);
            w += __shfl_xor(w, off, 32);
        }
        aL[t] = v; aR[t] = w;
    }
    if (lane == 0) {
#pragma unroll
        for (int t = 0; t < T_; ++t) {
            lhs[wid * T_ + t] = aL[t];
            rhs[wid * T_ + t] = aR[t];
        }
    }
}

// ---------------------------------------------------------------------------
// Stage 2: sig[b,i,j] = sigmoid( dot(lhs[b,i,:], rhs[b,j,:]) + bs[i,j] )
// One thread per output element; bandwidth bound (lhs/rhs live in L2).
// ---------------------------------------------------------------------------
__global__ void k_prod_sigmoid(const float* __restrict__ lhs,
                               const float* __restrict__ rhs,
                               const float* __restrict__ bs,
                               float* __restrict__ sig)
{
    const size_t idx = (size_t)blockIdx.x * blockDim.x + threadIdx.x; // < 16M
    const int b = (int)(idx >> 20);
    const int i = (int)((idx >> 10) & (N_ - 1));
    const int j = (int)(idx & (N_ - 1));
    const float* L = lhs + ((size_t)b * N_ + i) * T_;
    const float* R = rhs + ((size_t)b * N_ + j) * T_;
    float a = 0.f;
#pragma unroll
    for (int t = 0; t < T_; ++t) a += L[t] * R[t];
    a += bs[i * N_ + j];
    sig[idx] = 1.0f / (1.0f + __expf(-a));
}

// ---------------------------------------------------------------------------
// Stage 3: Spre[b,n,c] = sum_m Vs[n,m] * sig[b,m,c]   (fp32 WMMA 16x16x4)
// Wave tile: 32 n-rows x 64 c-cols -> 2 A-frags x 4 B-frags = 8 WMMAs/K-step.
// ---------------------------------------------------------------------------
__global__ void k_vs_gemm(const float* __restrict__ Vs,
                          const float* __restrict__ sig,
                          float* __restrict__ Spre)
{
    const int lane = threadIdx.x & 31;
    const int lm = lane & 15;
    const int lh = lane >> 4;                               // K-pair select
    const int gw = blockIdx.x * 8 + (threadIdx.x >> 5);     // 0..8191
    const int b  = gw >> 9;
    const int r  = gw & 511;
    const int n0 = (r >> 4) * 32;
    const int c0 = (r & 15) * 64;

    const float* arow0 = Vs + (size_t)(n0 + lm) * N_ + lh * 2;
    const float* arow1 = arow0 + (size_t)16 * N_;
    const float* sb    = sig + (size_t)b * N_ * N_ + c0 + lm;

    v8f acc[2][4];
    v8f zero = {};
#pragma unroll
    for (int v = 0; v < 2; ++v)
#pragma unroll
        for (int u = 0; u < 4; ++u) acc[v][u] = zero;

    for (int m = 0; m < N_; m += 4) {
        v2f a0 = *(const v2f*)(arow0 + m);                  // A[n0   +lm][m+lh*2 +{0,1}]
        v2f a1 = *(const v2f*)(arow1 + m);                  // A[n0+16+lm][m+lh*2 +{0,1}]
        const float* s0 = sb + (size_t)(m + lh * 2) * N_;
#pragma unroll
        for (int u = 0; u < 4; ++u) {
            v2f bf;
            bf.x = s0[u * 16];
            bf.y = s0[N_ + u * 16];
            acc[0][u] = wmma_f32_16x16x4(a0, bf, acc[0][u]);
            acc[1][u] = wmma_f32_16x16x4(a1, bf, acc[1][u]);
        }
    }
    float* dst = Spre + (size_t)b * N_ * N_;
#pragma unroll
    for (int v = 0; v < 2; ++v)
#pragma unroll
        for (int u = 0; u < 4; ++u)
#pragma unroll
            for (int rr = 0; rr < 8; ++rr)
                dst[(size_t)(n0 + v * 16 + rr + 8 * lh) * N_ + c0 + u * 16 + lm] =
                    acc[v][u][rr];
}

// ---------------------------------------------------------------------------
// Stage 4: softmax over axis=1 (over n), per (b, column c). Coalesced rows.
// ---------------------------------------------------------------------------
__global__ void k_softmax_col(float* __restrict__ S)
{
    const int gcol = blockIdx.x * 64 + threadIdx.x;         // 0..16383
    const int b = gcol >> 10;
    const int c = gcol & (N_ - 1);
    float* p = S + (size_t)b * N_ * N_ + c;
    float mx = -3.402823466e38f;
    for (int n = 0; n < N_; ++n) mx = fmaxf(mx, p[(size_t)n * N_]);
    float sum = 0.f;
    for (int n = 0; n < N_; ++n) sum += __expf(p[(size_t)n * N_] - mx);
    const float inv = 1.0f / sum;
    for (int n = 0; n < N_; ++n)
        p[(size_t)n * N_] = __expf(p[(size_t)n * N_] - mx) * inv;
}

// ---------------------------------------------------------------------------
// Stage 5 (per k): G[b,n,c] = sum_m cheb_k[m,n]*S[b,m,n] * x[b,m,c], c in [0,768)
// A-fragment built on the fly (elementwise cheb*S); fp32 WMMA, 32x64 wave tile.
// All operands L2-resident (S_b 4MB + x_b 3MB + cheb_k 4MB << 192MB).
// ---------------------------------------------------------------------------
__global__ void k_cheb_gemm(const float* __restrict__ chebk,
                            const float* __restrict__ S,
                            const float* __restrict__ x,
                            float* __restrict__ G)
{
    const int lane = threadIdx.x & 31;
    const int lm = lane & 15;
    const int lh = lane >> 4;
    const int gw = blockIdx.x * 8 + (threadIdx.x >> 5);     // 0..6143
    const int b  = gw / 384;
    const int r  = gw % 384;
    const int n0 = (r / 12) * 32;
    const int c0 = (r % 12) * 64;

    const float* ccol0 = chebk + n0 + lm;                          // + m*N_
    const float* scol0 = S + (size_t)b * N_ * N_ + n0 + lm;        // + m*N_
    const float* xb    = x + (size_t)b * N_ * FT_ + c0 + lm;       // + m*768

    v8f acc[2][4];
    v8f zero = {};
#pragma unroll
    for (int v = 0; v < 2; ++v)
#pragma unroll
        for (int u = 0; u < 4; ++u) acc[v][u] = zero;

    for (int m = 0; m < N_; m += 4) {
        const size_t k0 = (size_t)(m + lh * 2) * N_;
        v2f a0, a1;
        a0.x = ccol0[k0]           * scol0[k0];             // A'[n0+lm   ][m']
        a0.y = ccol0[k0 + N_]      * scol0[k0 + N_];
        a1.x = ccol0[k0 + 16]      * scol0[k0 + 16];        // A'[n0+16+lm][m']
        a1.y = ccol0[k0 + N_ + 16] * scol0[k0 + N_ + 16];
        const float* xr = xb + (size_t)(m + lh * 2) * FT_;
#pragma unroll
        for (int u = 0; u < 4; ++u) {
            v2f bf;
            bf.x = xr[u * 16];
            bf.y = xr[FT_ + u * 16];
            acc[0][u] = wmma_f32_16x16x4(a0, bf, acc[0][u]);
            acc[1][u] = wmma_f32_16x16x4(a1, bf, acc[1][u]);
        }
    }
    float* dst = G + (size_t)b * N_ * FT_;
#pragma unroll
    for (int v = 0; v < 2; ++v)
#pragma unroll
        for (int u = 0; u < 4; ++u)
#pragma unroll
            for (int rr = 0; rr < 8; ++rr)
                dst[(size_t)(n0 + v * 16 + rr + 8 * lh) * FT_ + c0 + u * 16 + lm] =
                    acc[v][u][rr];
}

// ---------------------------------------------------------------------------
// Stage 6 (per k): out[b,n,o,t] (+)= sum_f G[b,n,f,t] * Theta_k[f,o]; relu last.
// One wave per (b,n): D(64o x 12t) via 4 o-tiles, K-loop f by 4. Theta in L1.
// B columns t>=12 are zero-masked (no EXEC divergence around WMMA).
// ---------------------------------------------------------------------------
__global__ void k_theta(const float* __restrict__ G,
                        const float* __restrict__ Thk,
                        float* __restrict__ out,
                        int accum, int relu)
{
    const int lane = threadIdx.x & 31;
    const int lm = lane & 15;
    const int lh = lane >> 4;
    const int gw = blockIdx.x * 8 + (threadIdx.x >> 5);     // b*N + n
    const float* Gp = G + (size_t)gw * FT_;
    const int   tcol = (lm < T_) ? lm : (T_ - 1);
    const float msk  = (lm < T_) ? 1.0f : 0.0f;

    v8f acc[4];
    v8f zero = {};
#pragma unroll
    for (int u = 0; u < 4; ++u) acc[u] = zero;

    for (int f = 0; f < FIN_; f += 4) {
        const int fr = f + lh * 2;
        v2f bf;
        bf.x = Gp[fr * T_ + tcol] * msk;                    // B[f'][t]
        bf.y = Gp[(fr + 1) * T_ + tcol] * msk;
#pragma unroll
        for (int u = 0; u < 4; ++u) {
            v2f a;                                          // A[o][f'] = Theta[f'][o]
            a.x = Thk[fr * FOUT_ + u * 16 + lm];
            a.y = Thk[(fr + 1) * FOUT_ + u * 16 + lm];
            acc[u] = wmma_f32_16x16x4(a, bf, acc[u]);
        }
    }
    if (lm < T_) {
        float* op = out + (size_t)gw * (FOUT_ * T_) + lm;
#pragma unroll
        for (int u = 0; u < 4; ++u)
#pragma unroll
            for (int rr = 0; rr < 8; ++rr) {
                const int o = u * 16 + rr + 8 * lh;
                float v = acc[u][rr];
                float* p = op + o * T_;
                if (accum) v += *p;
                if (relu)  v = fmaxf(v, 0.f);
                *p = v;
            }
    }
}

// ---------------------------------------------------------------------------
extern "C" void kernel_launch(void* const* d_in, const int* in_sizes, int n_in,
                              void* d_out, int out_size, void* d_ws, size_t ws_size,
                              hipStream_t stream) {
    (void)in_sizes; (void)n_in; (void)out_size; (void)ws_size;
    const float* x    = (const float*)d_in[0];
    const float* W1   = (const float*)d_in[1];
    const float* W2   = (const float*)d_in[2];
    const float* W3   = (const float*)d_in[3];
    const float* bs   = (const float*)d_in[4];
    const float* Vs   = (const float*)d_in[5];
    const float* cheb = (const float*)d_in[6];
    const float* Th   = (const float*)d_in[7];
    float* out = (float*)d_out;

    // Workspace layout (floats): lhs | rhs | S | sig (sig region reused as G)
    float* ws  = (float*)d_ws;
    float* lhs = ws;
    float* rhs = lhs + (size_t)B_ * N_ * T_;
    float* Sm  = rhs + (size_t)B_ * N_ * T_;
    float* sig = Sm + (size_t)B_ * N_ * N_;
    float* G   = sig;   // 50.3 MB <= 67 MB, sig dead after stage 3

    k_attn_vecs<<<dim3((B_ * N_) / 8), dim3(256), 0, stream>>>(x, W1, W2, W3, lhs, rhs);
    k_prod_sigmoid<<<dim3((unsigned)((size_t)B_ * N_ * N_ / 256)), dim3(256), 0, stream>>>(
        lhs, rhs, bs, sig);
    k_vs_gemm<<<dim3((B_ * (N_ / 32) * (N_ / 64)) / 8), dim3(256), 0, stream>>>(Vs, sig, Sm);
    k_softmax_col<<<dim3((B_ * N_) / 64), dim3(64), 0, stream>>>(Sm);
    for (int k = 0; k < K_; ++k) {
        k_cheb_gemm<<<dim3((B_ * (N_ / 32) * (FT_ / 64)) / 8), dim3(256), 0, stream>>>(
            cheb + (size_t)k * N_ * N_, Sm, x, G);
        k_theta<<<dim3((B_ * N_) / 8), dim3(256), 0, stream>>>(
            G, Th + (size_t)k * FIN_ * FOUT_, out,
            /*accum=*/(k > 0) ? 1 : 0, /*relu=*/(k == K_ - 1) ? 1 : 0);
    }
}